// InfluencerLoss_11845519802383
// MI455X (gfx1250) — compile-verified
//
#include <hip/hip_runtime.h>
#include <hip/hip_bf16.h>
#include <math.h>

// MI455X / gfx1250, wave32. Gather-bound edge-distance loss; per-edge dist^2
// computed on the matrix pipe via V_WMMA_F32_16X16X4_F32 Gram diagonals (f32-exact).

typedef __attribute__((ext_vector_type(2))) float v2f;
typedef __attribute__((ext_vector_type(8))) float v8f;

#define D_DIM 128
#define WAVES_PER_BLOCK 4
#define EDGES_PER_WAVE 16
#define LDS_STRIDE 132  // 128 + 4 pad -> conflict-free WMMA operand reads

__global__ void zero_kernel(float* p, int n) {
    int i = blockIdx.x * blockDim.x + threadIdx.x;
    if (i < n) p[i] = 0.0f;
}

// MODE 0: UI-positive scatter  (atomicAdd dist^2 into follower_sum[dst], +1 into follower_cnt[dst])
// MODE 1: UI-negative hinge    (sum relu(1-d)^2 over truth==0, count negatives)
// MODE 2: II hinge             (sum relu(1-d)^2 over all edges)
template <int MODE>
__global__ void edge_kernel(const float* __restrict__ Q, const float* __restrict__ DB,
                            const int* __restrict__ e0, const int* __restrict__ e1,
                            const int* __restrict__ truth, int nE,
                            float* __restrict__ fol_sum, float* __restrict__ fol_cnt,
                            float* __restrict__ out_sum, float* __restrict__ out_cnt) {
    __shared__ float lds[WAVES_PER_BLOCK][EDGES_PER_WAVE][LDS_STRIDE];
    __shared__ float diag[WAVES_PER_BLOCK][EDGES_PER_WAVE];
    __shared__ float blk_sum;
    __shared__ float blk_cnt;

    const int lane = threadIdx.x & 31;
    const int wv   = threadIdx.x >> 5;
    const int waveg = blockIdx.x * WAVES_PER_BLOCK + wv;
    const int eb    = waveg * EDGES_PER_WAVE;     // wave-uniform base edge
    const bool active = (eb < nE);                // wave-uniform

    if (threadIdx.x == 0) { blk_sum = 0.0f; blk_cnt = 0.0f; }

    // ---- gather phase: diff rows (16 edges x 128 f32) into LDS, f32 exact ----
    if (active) {
        for (int i = 0; i < EDGES_PER_WAVE; ++i) {
            int e = eb + i;
            float4 d4 = make_float4(0.f, 0.f, 0.f, 0.f);
            if (e < nE) {
                int s = e0[e];
                int t = e1[e];
                const float4* qr = (const float4*)(Q  + (size_t)s * D_DIM);
                const float4* dr = (const float4*)(DB + (size_t)t * D_DIM);
                float4 a = qr[lane];   // 32 lanes x 16B = full 512B row
                float4 b = dr[lane];
                d4 = make_float4(a.x - b.x, a.y - b.y, a.z - b.z, a.w - b.w);
                // prefetch next edge's rows (global_prefetch_b8); compute is free, hide L2 latency
                if (i + 1 < EDGES_PER_WAVE && e + 1 < nE) {
                    int sn = e0[e + 1];
                    int tn = e1[e + 1];
                    __builtin_prefetch(Q  + (size_t)sn * D_DIM + lane * 4, 0, 3);
                    __builtin_prefetch(DB + (size_t)tn * D_DIM + lane * 4, 0, 3);
                }
            }
            *(float4*)&lds[wv][i][lane * 4] = d4;
        }
    }
    __syncthreads();  // unconditional: makes LDS diff visible; whole block participates

    // ---- WMMA phase: C += Diff(16x4) x Diff^T(4x16), 32 chunks => diag = dist^2 ----
    if (active) {
        v8f c = {};
        const int row = lane & 15;
        const int koff = (lane >> 4) << 1;  // +2 for upper lane half (A/B halving rule)
#pragma unroll 4
        for (int kc = 0; kc < 32; ++kc) {
            // A[m=row][k=vgpr+koff] == B[k=vgpr+koff][n=row] for a Gram => same operand
            v2f ab = *(const v2f*)&lds[wv][row][4 * kc + koff];
            c = __builtin_amdgcn_wmma_f32_16x16x4_f32(
                /*neg_a=*/false, ab, /*neg_b=*/false, ab,
                /*c_mod=*/(short)0, c, /*reuse_a=*/false, /*reuse_b=*/false);
        }
        // extract diagonal D[i][i]: i<8 -> lane i vgpr i ; i>=8 -> lane i+16 vgpr i-8
#pragma unroll
        for (int i = 0; i < 8; ++i)
            if (lane == i) diag[wv][i] = c[i];
#pragma unroll
        for (int i = 8; i < 16; ++i)
            if (lane == i + 16) diag[wv][i] = c[i - 8];
    }
    __syncthreads();

    // ---- per-edge consumption ----
    float partial = 0.0f, pcnt = 0.0f;
    if (active && lane < EDGES_PER_WAVE && (eb + lane) < nE) {
        int e = eb + lane;
        float dsq = diag[wv][lane];
        if (MODE == 0) {
            int t = e1[e];
            atomicAdd(&fol_sum[t], dsq);
            atomicAdd(&fol_cnt[t], 1.0f);
        } else if (MODE == 1) {
            if (truth[e] == 0) {  // hinge<0 <=> truth==0
                float d = sqrtf(dsq + 1e-12f);
                float r = fmaxf(1.0f - d, 0.0f);  // USER_MARGIN = 1
                partial = r * r;
                pcnt = 1.0f;
            }
        } else {
            float d = sqrtf(dsq + 1e-12f);
            float r = fmaxf(1.0f - d, 0.0f);      // INFL_MARGIN = 1
            partial = r * r;
        }
    }
    if (MODE != 0) {
        if (partial != 0.0f || pcnt != 0.0f) {
            atomicAdd(&blk_sum, partial);  // ds_add_f32
            if (MODE == 1) atomicAdd(&blk_cnt, pcnt);
        }
        __syncthreads();
        if (threadIdx.x == 0) {
            atomicAdd(out_sum, blk_sum);
            if (MODE == 1 && out_cnt) atomicAdd(out_cnt, blk_cnt);
        }
    }
}

__global__ void node_log_kernel(const float* __restrict__ fol_sum, const float* __restrict__ fol_cnt,
                                const int* __restrict__ pid, int n,
                                float* __restrict__ pid_acc, float* __restrict__ pid_cnt) {
    int i = blockIdx.x * blockDim.x + threadIdx.x;
    if (i < n) {
        float cnt = fmaxf(fol_cnt[i], 1.0f);
        float f = fol_sum[i] / cnt;     // / USER_MARGIN^2 (=1)
        float lf = logf(f);
        int g = pid[i];
        atomicAdd(&pid_acc[g], lf);
        atomicAdd(&pid_cnt[g], 1.0f);
    }
}

__global__ void group_kernel(const float* __restrict__ pid_acc, const float* __restrict__ pid_cnt,
                             int g_count, float* __restrict__ pos_sum) {
    __shared__ float s;
    if (threadIdx.x == 0) s = 0.0f;
    __syncthreads();
    int g = blockIdx.x * blockDim.x + threadIdx.x;
    if (g < g_count) {
        float c = fmaxf(pid_cnt[g], 1.0f);
        atomicAdd(&s, expf(pid_acc[g] / c));
    }
    __syncthreads();
    if (threadIdx.x == 0) atomicAdd(pos_sum, s);
}

__global__ void final_kernel(const float* __restrict__ scal, float* __restrict__ out,
                             float invG, float invE3) {
    // scal = [neg_sum, neg_cnt, ii_sum, pos_sum]
    const float W_UI = 1.0f, W_II = 1.0f, NEG_RATIO = 1.0f;
    float neg = scal[0] / fmaxf(scal[1], 1.0f);
    float pos = scal[3] * invG;
    float ii  = W_II * (scal[2] * invE3);
    float ui  = W_UI * (NEG_RATIO * neg + pos);
    out[0] = W_UI * ui + W_II * ii;
}

static inline int edge_blocks(int nE) {
    int waves = (nE + EDGES_PER_WAVE - 1) / EDGES_PER_WAVE;
    return (waves + WAVES_PER_BLOCK - 1) / WAVES_PER_BLOCK;
}

extern "C" void kernel_launch(void* const* d_in, const int* in_sizes, int n_in,
                              void* d_out, int out_size, void* d_ws, size_t ws_size,
                              hipStream_t stream) {
    const float* user = (const float*)d_in[0];
    const float* infl = (const float*)d_in[1];
    const int* pid       = (const int*)d_in[2];
    const int* edge_idx  = (const int*)d_in[3];
    const int* ui_edges  = (const int*)d_in[4];
    const int* ui_truth  = (const int*)d_in[5];
    const int* ii_edges  = (const int*)d_in[6];
    // d_in[7] = ii_truth (unused by 'hinge' type, matching reference)

    const int n  = in_sizes[0] / D_DIM;   // 100000
    const int E  = in_sizes[3] / 2;       // 400000
    const int E2 = in_sizes[4] / 2;       // 400000
    const int E3 = in_sizes[6] / 2;       // 400000
    const int G  = 1000;                  // unique pid count (reference constant)

    float* ws      = (float*)d_ws;
    float* fol_sum = ws;                  // [n]
    float* fol_cnt = ws + n;              // [n]
    float* pid_acc = ws + 2 * n;          // [G]
    float* pid_cnt = ws + 2 * n + G;      // [G]
    float* scal    = ws + 2 * n + 2 * G;  // [4]: neg_sum, neg_cnt, ii_sum, pos_sum
    int zn = 2 * n + 2 * G + 4;

    zero_kernel<<<(zn + 255) / 256, 256, 0, stream>>>(ws, zn);

    edge_kernel<0><<<edge_blocks(E), WAVES_PER_BLOCK * 32, 0, stream>>>(
        user, infl, edge_idx, edge_idx + E, nullptr, E,
        fol_sum, fol_cnt, nullptr, nullptr);

    edge_kernel<1><<<edge_blocks(E2), WAVES_PER_BLOCK * 32, 0, stream>>>(
        user, infl, ui_edges, ui_edges + E2, ui_truth, E2,
        nullptr, nullptr, &scal[0], &scal[1]);

    edge_kernel<2><<<edge_blocks(E3), WAVES_PER_BLOCK * 32, 0, stream>>>(
        infl, infl, ii_edges, ii_edges + E3, nullptr, E3,
        nullptr, nullptr, &scal[2], nullptr);

    node_log_kernel<<<(n + 255) / 256, 256, 0, stream>>>(fol_sum, fol_cnt, pid, n,
                                                         pid_acc, pid_cnt);
    group_kernel<<<(G + 255) / 256, 256, 0, stream>>>(pid_acc, pid_cnt, G, &scal[3]);
    final_kernel<<<1, 1, 0, stream>>>(scal, (float*)d_out, 1.0f / (float)G, 1.0f / (float)E3);
}